// GraphAttentionLayer_73804718014640
// MI455X (gfx1250) — compile-verified
//
#include <hip/hip_runtime.h>
#include <math.h>

#define NN   768   // nodes
#define INP  256   // input dim
#define HEADS 4
#define FH   64
#define OUTD (HEADS*FH)   // 256

typedef __attribute__((ext_vector_type(2))) float v2f;
typedef __attribute__((ext_vector_type(8))) float v8f;

// ---------------------------------------------------------------------------
// Kernel 1: Wx = x @ W^T   (768x256 = [768x256] @ [256x256]^T), fp32 WMMA.
// One wave (32 lanes) per 16x16 output tile; K loop in steps of 4 using
// V_WMMA_F32_16X16X4_F32.
// Lane layout (ISA 7.12.2, 32-bit):
//   A frag: a[v] = A[lane&15][k + 2*(lane>>4) + v]
//   B frag: b[v] = B[k + 2*(lane>>4) + v][lane&15]   (B = W^T, so B[k][o]=W[o][k])
//   C frag: c[v] = C[v + 8*(lane>>4)][lane&15]
// ---------------------------------------------------------------------------
__global__ __launch_bounds__(32)
void gat_gemm_xWt(const float* __restrict__ x, const float* __restrict__ W,
                  float* __restrict__ Wx)
{
    const int lane = threadIdx.x;
    const int col  = lane & 15;     // A row index == B/C col index
    const int half = lane >> 4;     // 0 or 1
    const int row0 = blockIdx.x * 16;   // node tile
    const int col0 = blockIdx.y * 16;   // output-feature tile

    const float* xr = x + (size_t)(row0 + col) * INP;  // A row for this lane
    const float* wr = W + (size_t)(col0 + col) * INP;  // W row == B column

    v8f c = {0.f, 0.f, 0.f, 0.f, 0.f, 0.f, 0.f, 0.f};

    for (int k = 0; k < INP; k += 4) {
        const int kk = k + 2 * half;          // even -> 8B aligned float2
        float2 av = *(const float2*)(xr + kk);
        float2 bv = *(const float2*)(wr + kk);
        v2f a, b;
        a[0] = av.x; a[1] = av.y;
        b[0] = bv.x; b[1] = bv.y;
        c = __builtin_amdgcn_wmma_f32_16x16x4_f32(
                /*neg_a=*/false, a, /*neg_b=*/false, b,
                /*c_mod=*/(short)0, c, /*reuse_a=*/false, /*reuse_b=*/false);
    }

#pragma unroll
    for (int v = 0; v < 8; ++v) {
        const int m = v + 8 * half;
        Wx[(size_t)(row0 + m) * OUTD + col0 + col] = c[v];
    }
}

// ---------------------------------------------------------------------------
// Kernel 2: pairwise scores + masked softmax.
// e[i,j,h] = sum_f a_f * leaky(Wx[i,h,f] + Wx[j,h,f]); |.| nonlinearity is not
// matmul-decomposable -> VALU. One block per (i,h); 256 threads cover j in 3
// strips of 256. Wx[i,h,:] and a[] cached in LDS; block-wide max/sum
// reductions in LDS (8 wave32 waves).
// ---------------------------------------------------------------------------
__global__ __launch_bounds__(256)
void gat_scores_softmax(const float* __restrict__ Wx, const int* __restrict__ adj,
                        const float* __restrict__ avec, float* __restrict__ att)
{
    const int i   = blockIdx.x;
    const int h   = blockIdx.y;
    const int tid = threadIdx.x;

    __shared__ float wxi[FH];
    __shared__ float af[FH];
    __shared__ float red[256];

    if (tid < FH) {
        wxi[tid] = Wx[(size_t)i * OUTD + h * FH + tid];
        af[tid]  = avec[tid];
    }
    __syncthreads();

    float ev[3];
#pragma unroll
    for (int r = 0; r < 3; ++r) {
        const int j = tid + r * 256;
        const float* wxj = Wx + (size_t)j * OUTD + h * FH;
        float s = 0.f;
#pragma unroll 8
        for (int f = 0; f < FH; ++f) {
            const float t = wxi[f] + wxj[f];
            // leaky_relu(t, 0.2) == max(t, 0.2t) for all t
            s = fmaf(af[f], fmaxf(t, 0.2f * t), s);
        }
        ev[r] = (adj[(size_t)i * NN + j] != 0) ? s : -INFINITY;
    }

    // block max over 768 scores
    float lm = fmaxf(fmaxf(ev[0], ev[1]), ev[2]);
    red[tid] = lm;
    __syncthreads();
    for (int s = 128; s > 0; s >>= 1) {
        if (tid < s) red[tid] = fmaxf(red[tid], red[tid + s]);
        __syncthreads();
    }
    const float mx = red[0];
    __syncthreads();

    float ex[3];
    float le = 0.f;
#pragma unroll
    for (int r = 0; r < 3; ++r) {
        ex[r] = __expf(ev[r] - mx);
        le += ex[r];
    }
    red[tid] = le;
    __syncthreads();
    for (int s = 128; s > 0; s >>= 1) {
        if (tid < s) red[tid] += red[tid + s];
        __syncthreads();
    }
    const float inv = 1.f / red[0];

    float* arow = att + ((size_t)h * NN + i) * NN;
#pragma unroll
    for (int r = 0; r < 3; ++r)
        arow[tid + r * 256] = ex[r] * inv;
}

// ---------------------------------------------------------------------------
// Kernel 3: per-head aggregation  out[i, h*64+f] = sum_j att[h][i][j]*Wx[j, h*64+f]
// A = att[h] (768x768), B = Wx head slice (768x64), fp32 WMMA, K stepped by 4.
// grid = (48 M-tiles, 4 heads, 4 N-tiles-within-head), one wave per tile.
// ---------------------------------------------------------------------------
__global__ __launch_bounds__(32)
void gat_gemm_aggregate(const float* __restrict__ att, const float* __restrict__ Wx,
                        float* __restrict__ out)
{
    const int lane = threadIdx.x;
    const int col  = lane & 15;
    const int half = lane >> 4;
    const int row0 = blockIdx.x * 16;   // node tile
    const int h    = blockIdx.y;        // head
    const int col0 = blockIdx.z * 16;   // feature tile within head (0..48)

    const float* A  = att + (size_t)h * NN * NN + (size_t)(row0 + col) * NN;
    const float* Bc = Wx + h * FH + col0 + col;   // column of B, stride OUTD

    v8f c = {0.f, 0.f, 0.f, 0.f, 0.f, 0.f, 0.f, 0.f};

    for (int k = 0; k < NN; k += 4) {
        const int kk = k + 2 * half;
        float2 av = *(const float2*)(A + kk);
        v2f a, b;
        a[0] = av.x; a[1] = av.y;
        b[0] = Bc[(size_t)(kk + 0) * OUTD];
        b[1] = Bc[(size_t)(kk + 1) * OUTD];
        c = __builtin_amdgcn_wmma_f32_16x16x4_f32(
                false, a, false, b, (short)0, c, false, false);
    }

#pragma unroll
    for (int v = 0; v < 8; ++v) {
        const int m = v + 8 * half;
        out[(size_t)(row0 + m) * OUTD + h * FH + col0 + col] = c[v];
    }
}

// ---------------------------------------------------------------------------
// Launcher. Inputs (setup_inputs order): x [768*256] f32, adj_mat [768*768*1]
// i32, W [256*256] f32, a [64] f32. Output: [768*256] f32.
// Workspace: Wx (768*256 f32) then att (4*768*768 f32)  ~= 9.8 MB.
// ---------------------------------------------------------------------------
extern "C" void kernel_launch(void* const* d_in, const int* in_sizes, int n_in,
                              void* d_out, int out_size, void* d_ws, size_t ws_size,
                              hipStream_t stream) {
    (void)in_sizes; (void)n_in; (void)out_size; (void)ws_size;

    const float* x   = (const float*)d_in[0];
    const int*   adj = (const int*)d_in[1];
    const float* W   = (const float*)d_in[2];
    const float* a   = (const float*)d_in[3];
    float*       out = (float*)d_out;

    float* Wx  = (float*)d_ws;                    // 768*256 floats
    float* att = Wx + (size_t)NN * OUTD;          // 4*768*768 floats

    gat_gemm_xWt<<<dim3(NN / 16, OUTD / 16), 32, 0, stream>>>(x, W, Wx);
    gat_scores_softmax<<<dim3(NN, HEADS), 256, 0, stream>>>(Wx, adj, a, att);
    gat_gemm_aggregate<<<dim3(NN / 16, HEADS, FH / 16), 32, 0, stream>>>(att, Wx, out);
}